// MultiHeadAttention_88854283419963
// MI455X (gfx1250) — compile-verified
//
#include <hip/hip_runtime.h>

// ---------------- problem constants (match reference) ----------------
#define Bsz   2
#define Lsz   128
#define Dsz   768
#define Hsz   4
#define DKsz  192
#define HDsz  384
#define HD1sz 385
#define NCsz  2
#define JCsz  16                 // j-values per chunk
#define COLSsz 32                // JCsz * NCsz
#define NCHsz 25                 // ceil(385/16)

typedef __attribute__((ext_vector_type(2))) float v2f;
typedef __attribute__((ext_vector_type(8))) float v8f;
typedef __attribute__((ext_vector_type(4))) unsigned int u32x4;
typedef __attribute__((ext_vector_type(8))) int i32x8;
typedef __attribute__((ext_vector_type(4))) int i32x4;

// CDNA5 fp32 matrix core: D(16x16,f32) = A(16x4,f32) * B(4x16,f32) + C
__device__ __forceinline__ v8f wmma4(v2f a, v2f b, v8f c) {
  return __builtin_amdgcn_wmma_f32_16x16x4_f32(false, a, false, b, (short)0, c,
                                               false, false);
}

#if __has_builtin(__builtin_amdgcn_tensor_load_to_lds) && \
    __has_builtin(__builtin_amdgcn_s_wait_tensorcnt)
#define USE_TDM 1
#else
#define USE_TDM 0
#endif

#if USE_TDM
// ---------------------------------------------------------------------
// Tensor Data Mover: DMA a 2D f32 tile (nrows x rowlen, row stride
// rowstride floats) from global memory into LDS at lds_byte.
// D# per CDNA5 ISA 8.3/8.4: group0 = {count=1, lds_addr, global_addr,
// type=2}; group1 = {data_size=4B, tensor_dim0/1, tile_dim0/1, stride};
// groups 2/3 zero (<=2D tensor). Issue from ONE wave; TENSORcnt tracked.
// 6-arg builtin form (clang-23 / therock-10.0 headers): the trailing
// int32x8 carries the remaining (unused) descriptor words -> zeros.
// ---------------------------------------------------------------------
__device__ __forceinline__ void tdm_load_tile_f32(const float* gsrc,
                                                  unsigned lds_byte,
                                                  unsigned rowlen,
                                                  unsigned rowstride,
                                                  unsigned nrows) {
  const unsigned long long ga = (unsigned long long)(size_t)gsrc;
  u32x4 g0;
  g0[0] = 1u;                                   // count=1, user mode
  g0[1] = lds_byte;                             // lds_addr (bytes)
  g0[2] = (unsigned)(ga & 0xFFFFFFFFu);         // global_addr[31:0]
  g0[3] = (unsigned)((ga >> 32) & 0x01FFFFFFu)  // global_addr[56:32]
          | (2u << 30);                         // type = 2 ("image")
  i32x8 g1;
  g1[0] = (int)0x00020000u;                     // data_size = 4 bytes
  g1[1] = (int)((rowlen & 0xFFFFu) << 16);      // tensor_dim0[15:0]
  g1[2] = (int)((rowlen >> 16) | ((nrows & 0xFFFFu) << 16));  // dim0 hi | dim1 lo
  g1[3] = (int)((nrows >> 16) | ((rowlen & 0xFFFFu) << 16));  // dim1 hi | tile_dim0
  g1[4] = (int)(nrows & 0xFFFFu);               // tile_dim1 (tile_dim2 = 0)
  g1[5] = (int)rowstride;                       // tensor_dim0_stride[31:0]
  g1[6] = 0;                                    // stride hi | dim1_stride lo
  g1[7] = 0;
  i32x4 g2 = {0, 0, 0, 0}, g3 = {0, 0, 0, 0};
  i32x8 g4 = {0, 0, 0, 0, 0, 0, 0, 0};
  __builtin_amdgcn_tensor_load_to_lds(g0, g1, g2, g3, g4, 0);
}
#endif

// Stage nrows x rowlen f32 tile (global row stride = rowstride) into LDS.
// Must be called by ALL threads of the block; ends with a barrier.
__device__ __forceinline__ void stage_tile(float* lds, const float* gsrc,
                                           int rowlen, int rowstride,
                                           int nrows) {
#if USE_TDM
  if (threadIdx.x == 0)  // waves with EXEC==0 branch around the TDM issue
    tdm_load_tile_f32(gsrc, (unsigned)(size_t)lds, (unsigned)rowlen,
                      (unsigned)rowstride, (unsigned)nrows);
  __builtin_amdgcn_s_wait_tensorcnt(0);
#else
  const int q = rowlen >> 2;
  for (int t = threadIdx.x; t < nrows * q; t += (int)blockDim.x) {
    const int r = t / q, c4 = t % q;
    *(float4*)(lds + r * rowlen + c4 * 4) =
        *(const float4*)(gsrc + (size_t)r * rowstride + c4 * 4);
  }
#endif
  __syncthreads();
}

// ---------------------------------------------------------------------
// Generic fp32 GEMM: C[M,N] = A[M,K](lda) * B[K,N](ldb) + bias, opt relu
// REQUIRES: M%16==0, N%64==0, K%4==0 (true for every call site here).
// block = 128 threads (4 waves); wave = one 16x16 tile; block = 16x64.
// A tile (16 x K) staged in LDS once (TDM), shared by all 4 waves.
// grid = (N/64, M/16)
// ---------------------------------------------------------------------
__global__ void gemm_bias_kernel(const float* __restrict__ A, int lda,
                                 const float* __restrict__ Bm, int ldb,
                                 const float* __restrict__ bias,
                                 float* __restrict__ C, int ldc,
                                 int M, int N, int K, int do_relu) {
  __shared__ float As[16 * Dsz];  // max K = 768 -> 48KB
  const int wave = threadIdx.x >> 5, lane = threadIdx.x & 31;
  const int hf = lane >> 4, l16 = lane & 15;
  const int m0 = blockIdx.y * 16;
  const int n0 = blockIdx.x * 64 + wave * 16;
  stage_tile(As, A + (size_t)m0 * lda, K, lda, 16);
  const float* arow = As + l16 * K + 2 * hf;
  const float* bcol = Bm + (size_t)2 * hf * ldb + n0 + l16;
  v8f c = {0.f, 0.f, 0.f, 0.f, 0.f, 0.f, 0.f, 0.f};
#pragma unroll 4
  for (int k0 = 0; k0 < K; k0 += 4) {
    v2f a = *(const v2f*)(arow + k0);  // ds_load_b64
    v2f bb;
    bb.x = bcol[(size_t)k0 * ldb];
    bb.y = bcol[(size_t)(k0 + 1) * ldb];
    c = wmma4(a, bb, c);
  }
  const int col = n0 + l16;
  const float bv = bias[col];
  for (int v = 0; v < 8; ++v) {
    float val = c[v] + bv;
    if (do_relu && val < 0.f) val = 0.f;
    C[(size_t)(m0 + v + 8 * hf) * ldc + col] = val;
  }
}

// ---------------------------------------------------------------------
// MHA raw scores: scores[b,h,q,k] = dot(qp[b,q,h*DK..], kp[b,k,h*DK..])
// Both fragments are contiguous along d -> b64 loads. grid = (2,8,B*H)
// ---------------------------------------------------------------------
__global__ void attn_scores_kernel(const float* __restrict__ qp,
                                   const float* __restrict__ kp,
                                   float* __restrict__ scores) {
  const int wave = threadIdx.x >> 5, lane = threadIdx.x & 31;
  const int hf = lane >> 4, l16 = lane & 15;
  const int bh = blockIdx.z, b = bh >> 2, h = bh & 3;
  const int m0 = blockIdx.y * 16;
  const int n0 = blockIdx.x * 64 + wave * 16;
  const float* qrow = qp + (size_t)(b * Lsz + m0 + l16) * Dsz + h * DKsz + 2 * hf;
  const float* krow = kp + (size_t)(b * Lsz + n0 + l16) * Dsz + h * DKsz + 2 * hf;
  v8f c = {0.f, 0.f, 0.f, 0.f, 0.f, 0.f, 0.f, 0.f};
#pragma unroll 4
  for (int k0 = 0; k0 < DKsz; k0 += 4) {
    v2f a = *(const v2f*)(qrow + k0);
    v2f bb = *(const v2f*)(krow + k0);
    c = wmma4(a, bb, c);
  }
  float* orow = scores + (size_t)(b * Hsz + h) * Lsz * Lsz;
  for (int v = 0; v < 8; ++v)
    orow[(size_t)(m0 + v + 8 * hf) * Lsz + n0 + l16] = c[v];
}

// scale + mask(-1e9) + softmax over k; one wave per (b,h,q) row, in place
__global__ void attn_softmax_kernel(float* __restrict__ scores,
                                    const int* __restrict__ mask) {
  const int lane = threadIdx.x & 31;
  const int row = (blockIdx.x * blockDim.x + threadIdx.x) >> 5;  // 0..1023
  const int b = row >> 9;  // H*L = 512
  const int q = row & 127;
  float* p = scores + (size_t)row * Lsz;
  const int* mrow = mask + (size_t)(b * Lsz + q) * Lsz;
  const float scale = 0.07216878364870323f;  // 1/sqrt(192)
  float vals[4];
  float mx = -3.4e38f;
  for (int t = 0; t < 4; ++t) {
    const int k = lane + 32 * t;
    float v = p[k] * scale;
    if (mrow[k] == 0) v = -1e9f;
    vals[t] = v;
    mx = fmaxf(mx, v);
  }
  for (int off = 16; off >= 1; off >>= 1) mx = fmaxf(mx, __shfl_xor(mx, off, 32));
  float sum = 0.f;
  for (int t = 0; t < 4; ++t) { vals[t] = expf(vals[t] - mx); sum += vals[t]; }
  for (int off = 16; off >= 1; off >>= 1) sum += __shfl_xor(sum, off, 32);
  const float inv = 1.f / sum;
  for (int t = 0; t < 4; ++t) p[lane + 32 * t] = vals[t] * inv;
}

// ---------------------------------------------------------------------
// D1: Tc[b,z,i,c] = sum_k mid[b,z,k] * Wtri[i,k,j0+c/2,c%2]
// A tile (16 z-rows x 384) DMA'd to LDS; invalid columns handled with a
// clamped base pointer + x{0,1} mask (loads stay unconditional).
// Wtri column walk strides 3080B -> prefetch 16 k-steps ahead.
// grid = (193, 8, B)
// ---------------------------------------------------------------------
__global__ void tri_d1_kernel(const float* __restrict__ mid,
                              const float* __restrict__ Wtri,
                              float* __restrict__ Tc, int j0) {
  __shared__ float As[16 * HDsz];  // 24KB
  const int wave = threadIdx.x >> 5, lane = threadIdx.x & 31;
  const int hf = lane >> 4, l16 = lane & 15;
  const int b = blockIdx.z;
  const int z0 = blockIdx.y * 16;
  const int n0 = blockIdx.x * 64 + wave * 16;
  stage_tile(As, mid + (size_t)(b * Lsz + z0) * HDsz, HDsz, HDsz, 16);
  const int n = n0 + l16;
  const bool nok = n < HD1sz * COLSsz;
  const int i = nok ? (n / COLSsz) : 0;
  const int cidx = nok ? (n % COLSsz) : 0;
  const int j = j0 + (cidx >> 1);
  const int r = cidx & 1;
  const bool jok = nok && (j < HD1sz);
  const float msk = jok ? 1.f : 0.f;
  const size_t kstr = (size_t)HD1sz * NCsz;  // 770
  const float* wb = Wtri +
      (jok ? ((size_t)i * HDsz * kstr + (size_t)j * NCsz + r) : 0) +
      (size_t)2 * hf * kstr;
  const float* arow = As + l16 * HDsz + 2 * hf;
  v8f c = {0.f, 0.f, 0.f, 0.f, 0.f, 0.f, 0.f, 0.f};
#pragma unroll 4
  for (int k0 = 0; k0 < HDsz; k0 += 4) {
    if ((k0 & 15) == 0)
      __builtin_prefetch(wb + (size_t)(k0 + 16) * kstr, 0, 1);
    v2f a = *(const v2f*)(arow + k0);  // ds_load_b64
    v2f bb;
    bb.x = wb[(size_t)k0 * kstr] * msk;
    bb.y = wb[(size_t)(k0 + 1) * kstr] * msk;
    c = wmma4(a, bb, c);
  }
  if (nok) {
    for (int v = 0; v < 8; ++v) {
      const int z = z0 + v + 8 * hf;
      Tc[((size_t)(b * Lsz + z) * HD1sz + i) * COLSsz + cidx] = c[v];
    }
  }
}

// ---------------------------------------------------------------------
// D2: Uc[b,x,z,c] = sum_i X[b,x,i]*Tc[b,z,i,c]; X=[head,1]
// Main loop i in [0,384) unconditional (head tile DMA'd to LDS); i==384
// bias row handled as one epilogue WMMA step. grid = (2, 8, B*COLS)
// ---------------------------------------------------------------------
__global__ void tri_d2_kernel(const float* __restrict__ head,
                              const float* __restrict__ Tc,
                              float* __restrict__ Uc) {
  __shared__ float As[16 * HDsz];  // 24KB
  const int wave = threadIdx.x >> 5, lane = threadIdx.x & 31;
  const int hf = lane >> 4, l16 = lane & 15;
  const int bc = blockIdx.z;
  const int b = bc / COLSsz, cidx = bc % COLSsz;
  const int m0 = blockIdx.y * 16;              // x
  const int n0 = blockIdx.x * 64 + wave * 16;  // z
  stage_tile(As, head + (size_t)(b * Lsz + m0) * HDsz, HDsz, HDsz, 16);
  const float* arow = As + l16 * HDsz + 2 * hf;
  const float* tcb = Tc + (size_t)(b * Lsz + n0 + l16) * HD1sz * COLSsz + cidx +
                     (size_t)2 * hf * COLSsz;
  v8f c = {0.f, 0.f, 0.f, 0.f, 0.f, 0.f, 0.f, 0.f};
#pragma unroll 4
  for (int k0 = 0; k0 < HDsz; k0 += 4) {
    v2f a = *(const v2f*)(arow + k0);
    v2f bb;
    bb.x = tcb[(size_t)k0 * COLSsz];
    bb.y = tcb[(size_t)(k0 + 1) * COLSsz];
    c = wmma4(a, bb, c);
  }
  {  // epilogue: i = 384 (bias row of X): A = 1 on lanes hf==0, else 0
    const float m = hf ? 0.f : 1.f;
    v2f a; a.x = m; a.y = 0.f;
    v2f bb;
    // clamped in-bounds load of Tc row 384, masked for hf==1
    bb.x = Tc[((size_t)(b * Lsz + n0 + l16) * HD1sz + HDsz) * COLSsz + cidx] * m;
    bb.y = 0.f;
    c = wmma4(a, bb, c);
  }
  for (int v = 0; v < 8; ++v) {
    const int x = m0 + v + 8 * hf;
    Uc[((size_t)(b * Lsz + x) * Lsz + n0 + l16) * COLSsz + cidx] = c[v];
  }
}

// ---------------------------------------------------------------------
// D3: s[b,x,y,z,r] (+)= sum_j Y[b,y,j]*Uc[b,x,z,(j-j0)*2+r]; Y=[tail,1]
// GEMM per (b,x): M=y=128, N=z*2+r=256, K=16. Full chunks (j0+16<=384):
// fully unconditional loads. Tail chunk (j0=384): A is pure constants.
// Accumulation via WMMA C operand. grid = (4, 8, B*L)
// ---------------------------------------------------------------------
__global__ void tri_d3_kernel(const float* __restrict__ tail,
                              const float* __restrict__ Uc,
                              float* __restrict__ s, int j0, int accum) {
  const int wave = threadIdx.x >> 5, lane = threadIdx.x & 31;
  const int hf = lane >> 4, l16 = lane & 15;
  const int bx = blockIdx.z;
  const int b = bx >> 7, x = bx & 127;
  const int m0 = blockIdx.y * 16;              // y
  const int n0 = blockIdx.x * 64 + wave * 16;  // z*NC + r
  const int n = n0 + l16;
  const int zc = n >> 1, rr = n & 1;
  float* sb = s + (size_t)(b * Lsz + x) * Lsz * Lsz * NCsz;
  v8f c;
  if (accum) {
    for (int v = 0; v < 8; ++v)
      c[v] = sb[(size_t)(m0 + v + 8 * hf) * (Lsz * NCsz) + n];
  } else {
    for (int v = 0; v < 8; ++v) c[v] = 0.f;
  }
  const float* urow = Uc + ((size_t)(b * Lsz + x) * Lsz + zc) * COLSsz + rr +
                      (size_t)2 * hf * NCsz;
  if (j0 + JCsz <= HDsz) {  // full chunk: all j < 384
    const float* trow = tail + (size_t)(b * Lsz + m0 + l16) * HDsz + j0 + 2 * hf;
#pragma unroll
    for (int k0 = 0; k0 < JCsz; k0 += 4) {
      v2f a = *(const v2f*)(trow + k0);  // contiguous b64
      v2f bb;
      bb.x = urow[(size_t)k0 * NCsz];
      bb.y = urow[(size_t)(k0 + 1) * NCsz];
      c = wmma4(a, bb, c);
    }
  } else {  // tail chunk (j0 == 384): Y = 1 at j==384, else 0
#pragma unroll
    for (int k0 = 0; k0 < JCsz; k0 += 4) {
      const int j = j0 + k0 + 2 * hf;
      v2f a;
      a.x = (j == HDsz) ? 1.f : 0.f;
      a.y = (j + 1 == HDsz) ? 1.f : 0.f;
      v2f bb;  // always in-bounds; A==0 kills garbage contributions
      bb.x = urow[(size_t)k0 * NCsz];
      bb.y = urow[(size_t)(k0 + 1) * NCsz];
      c = wmma4(a, bb, c);
    }
  }
  for (int v = 0; v < 8; ++v)
    sb[(size_t)(m0 + v + 8 * hf) * (Lsz * NCsz) + n] = c[v];
}

// mask (z>y || z<x -> -1e6) + softmax over z; one wave per (b,x,y,r)
__global__ void tri_softmax_z_kernel(float* __restrict__ s) {
  const int lane = threadIdx.x & 31;
  int w = (blockIdx.x * blockDim.x + threadIdx.x) >> 5;  // 0..65535
  const int r = w & 1; w >>= 1;
  const int y = w & 127; w >>= 7;
  const int x = w & 127; const int b = w >> 7;
  float* p = s + ((size_t)(b * Lsz + x) * Lsz + y) * (Lsz * NCsz) + r;
  float vals[4];
  float mx = -3.4e38f;
  for (int t = 0; t < 4; ++t) {
    const int z = lane + 32 * t;
    float v = p[(size_t)z * NCsz];
    if (z > y || z < x) v = -1e6f;
    vals[t] = v;
    mx = fmaxf(mx, v);
  }
  for (int off = 16; off >= 1; off >>= 1) mx = fmaxf(mx, __shfl_xor(mx, off, 32));
  float sum = 0.f;
  for (int t = 0; t < 4; ++t) { vals[t] = expf(vals[t] - mx); sum += vals[t]; }
  for (int off = 16; off >= 1; off >>= 1) sum += __shfl_xor(sum, off, 32);
  const float inv = 1.f / sum;
  for (int t = 0; t < 4; ++t) p[(size_t)(lane + 32 * t) * NCsz] = vals[t] * inv;
}

// ---------------------------------------------------------------------
// F: score[b,i,j,r] = sum_h relu(sum_z alpha[b,i,j,z,r]*text[b,z,h])*Vw[r,h]+Vb[r]
// alpha tile (16 rows x 256) DMA'd to LDS, shared by the 4 waves.
// grid = (1024, B*NC), block 128; deterministic staged LDS reduction.
// ---------------------------------------------------------------------
__global__ void tri_score_kernel(const float* __restrict__ alpha,
                                 const float* __restrict__ text,
                                 const float* __restrict__ Vw,
                                 const float* __restrict__ Vb,
                                 float* __restrict__ score) {
  __shared__ float As[16 * Lsz * NCsz];  // 16KB
  __shared__ float rowsum[4][16];
  const int wave = threadIdx.x >> 5, lane = threadIdx.x & 31;
  const int hf = lane >> 4, l16 = lane & 15;
  const int br = blockIdx.y, b = br >> 1, r = br & 1;
  const int m0 = blockIdx.x * 16;  // flattened row i*L + j
  stage_tile(As, alpha + ((size_t)b * Lsz * Lsz + m0) * 256, 256, 256, 16);
  const float* arow = As + l16 * 256 + r + 4 * hf;  // z index stride 2
  float partial[8] = {0.f, 0.f, 0.f, 0.f, 0.f, 0.f, 0.f, 0.f};
  for (int nt = wave; nt < Dsz / 16; nt += 4) {
    const int n0 = nt * 16;
    const float* brow = text + ((size_t)b * Lsz + 2 * hf) * Dsz + n0 + l16;
    v8f c = {0.f, 0.f, 0.f, 0.f, 0.f, 0.f, 0.f, 0.f};
#pragma unroll 4
    for (int k0 = 0; k0 < Lsz; k0 += 4) {
      v2f a;
      a.x = arow[(size_t)k0 * NCsz];
      a.y = arow[(size_t)k0 * NCsz + 2];
      v2f bb;
      bb.x = brow[(size_t)k0 * Dsz];
      bb.y = brow[(size_t)(k0 + 1) * Dsz];
      c = wmma4(a, bb, c);
    }
    const float vw = Vw[r * Dsz + n0 + l16];
    for (int v = 0; v < 8; ++v) partial[v] += fmaxf(c[v], 0.f) * vw;
  }
  for (int off = 8; off >= 1; off >>= 1)
    for (int v = 0; v < 8; ++v) partial[v] += __shfl_xor(partial[v], off, 32);
  if (l16 == 0)
    for (int v = 0; v < 8; ++v) rowsum[wave][v + 8 * hf] = partial[v];
  __syncthreads();
  if (threadIdx.x < 16) {
    const float tot = rowsum[0][threadIdx.x] + rowsum[1][threadIdx.x] +
                      rowsum[2][threadIdx.x] + rowsum[3][threadIdx.x];
    const int ro = m0 + threadIdx.x;
    score[((size_t)b * Lsz * Lsz + ro) * NCsz + r] = tot + Vb[r];
  }
}

// pad + raw-reshape [B,L,L,4]->[B,4,L,L] + add p_attn
__global__ void comb_kernel(const float* __restrict__ pattn,
                            const float* __restrict__ score,
                            float* __restrict__ comb) {
  const int idx = blockIdx.x * blockDim.x + threadIdx.x;  // 0..131071
  const int b = idx >> 16;
  const int f = idx & 65535;
  const int i = f >> 9;
  const int j = (f >> 2) & 127;
  const int cch = f & 3;
  float sc = 0.f;
  if (cch < NCsz)
    sc = score[((size_t)b * Lsz * Lsz + i * Lsz + j) * NCsz + cch];
  comb[idx] = pattn[idx] + sc;
}

// global min / max (single block, deterministic tree)
__global__ void minmax_kernel(const float* __restrict__ comb,
                              float* __restrict__ mm) {
  __shared__ float slo[256], shi[256];
  float lo = 3.4e38f, hi = -3.4e38f;
  for (int i = threadIdx.x; i < Bsz * Hsz * Lsz * Lsz; i += 256) {
    const float v = comb[i];
    lo = fminf(lo, v);
    hi = fmaxf(hi, v);
  }
  slo[threadIdx.x] = lo;
  shi[threadIdx.x] = hi;
  __syncthreads();
  for (int st = 128; st >= 1; st >>= 1) {
    if (threadIdx.x < st) {
      slo[threadIdx.x] = fminf(slo[threadIdx.x], slo[threadIdx.x + st]);
      shi[threadIdx.x] = fmaxf(shi[threadIdx.x], shi[threadIdx.x + st]);
    }
    __syncthreads();
  }
  if (threadIdx.x == 0) { mm[0] = slo[0]; mm[1] = shi[0]; }
}

// normalize + softmax over k; one wave per (b,h,q) row
__global__ void final_softmax_kernel(const float* __restrict__ comb,
                                     const float* __restrict__ mm,
                                     float* __restrict__ out) {
  const int lane = threadIdx.x & 31;
  const int row = (blockIdx.x * blockDim.x + threadIdx.x) >> 5;  // 0..1023
  const float lo = mm[0];
  const float inv = 1.f / (mm[1] - mm[0]);
  const float* p = comb + (size_t)row * Lsz;
  float vals[4];
  float mx = -3.4e38f;
  for (int t = 0; t < 4; ++t) {
    const float v = (p[lane + 32 * t] - lo) * inv;
    vals[t] = v;
    mx = fmaxf(mx, v);
  }
  for (int off = 16; off >= 1; off >>= 1) mx = fmaxf(mx, __shfl_xor(mx, off, 32));
  float sum = 0.f;
  for (int t = 0; t < 4; ++t) { vals[t] = expf(vals[t] - mx); sum += vals[t]; }
  for (int off = 16; off >= 1; off >>= 1) sum += __shfl_xor(sum, off, 32);
  const float isum = 1.f / sum;
  float* o = out + (size_t)row * Lsz;
  for (int t = 0; t < 4; ++t) o[lane + 32 * t] = vals[t] * isum;
}

// ---------------------------------------------------------------------
extern "C" void kernel_launch(void* const* d_in, const int* in_sizes, int n_in,
                              void* d_out, int out_size, void* d_ws,
                              size_t ws_size, hipStream_t stream) {
  const float* query = (const float*)d_in[0];
  const float* key   = (const float*)d_in[1];
  const float* text  = (const float*)d_in[2];
  const int*   mask  = (const int*)d_in[3];
  const float* Wq = (const float*)d_in[4];  const float* bq = (const float*)d_in[5];
  const float* Wk = (const float*)d_in[6];  const float* bk = (const float*)d_in[7];
  const float* W1s[3] = {(const float*)d_in[8],  (const float*)d_in[12], (const float*)d_in[16]};
  const float* b1s[3] = {(const float*)d_in[9],  (const float*)d_in[13], (const float*)d_in[17]};
  const float* W2s[3] = {(const float*)d_in[10], (const float*)d_in[14], (const float*)d_in[18]};
  const float* b2s[3] = {(const float*)d_in[11], (const float*)d_in[15], (const float*)d_in[19]};
  const float* Wtri = (const float*)d_in[20];
  const float* Vw   = (const float*)d_in[21];
  const float* Vb   = (const float*)d_in[22];

  float* ws = (float*)d_ws;
  // workspace layout (floats), total ~54.8 MB
  float* qp     = ws;                         // 196608
  float* kp     = qp + 196608;                // 196608
  float* pattn  = kp + 196608;                // 131072 (scores -> p_attn)
  float* headb  = pattn + 131072;             // 98304
  float* tailb  = headb + 98304;              // 98304
  float* midb   = tailb + 98304;              // 98304
  float* hidden = midb + 98304;               // 98304 (reused per MLP)
  float* scoreb = hidden + 98304;             // 65536
  float* comb   = scoreb + 65536;             // 131072
  float* mm     = comb + 131072;              // 2 (+pad)
  float* sbuf   = mm + 4;                     // 8388608  [B,L,L,L,NC]
  float* Tc     = sbuf + 8388608;             // 3153920
  float* Uc     = Tc + 3153920;               // 1048576

  const dim3 blk(128);
  float* mlp_out[3] = {headb, tailb, midb};

  // A: q/k projections  [256,768] x [768,768]
  gemm_bias_kernel<<<dim3(12, 16), blk, 0, stream>>>(query, Dsz, Wq, Dsz, bq,
                                                     qp, Dsz, Bsz * Lsz, Dsz, Dsz, 0);
  gemm_bias_kernel<<<dim3(12, 16), blk, 0, stream>>>(key, Dsz, Wk, Dsz, bk,
                                                     kp, Dsz, Bsz * Lsz, Dsz, Dsz, 0);
  // B: scores + masked softmax -> p_attn
  attn_scores_kernel<<<dim3(2, 8, Bsz * Hsz), blk, 0, stream>>>(qp, kp, pattn);
  attn_softmax_kernel<<<dim3(256), blk, 0, stream>>>(pattn, mask);

  // C: head / tail / mid MLPs
  for (int m = 0; m < 3; ++m) {
    gemm_bias_kernel<<<dim3(6, 16), blk, 0, stream>>>(text, Dsz, W1s[m], HDsz,
                                                      b1s[m], hidden, HDsz,
                                                      Bsz * Lsz, HDsz, Dsz, 1);
    gemm_bias_kernel<<<dim3(6, 16), blk, 0, stream>>>(hidden, HDsz, W2s[m], HDsz,
                                                      b2s[m], mlp_out[m], HDsz,
                                                      Bsz * Lsz, HDsz, HDsz, 0);
  }

  // D: tri-affine chain, chunked over (j,r) columns
  for (int c = 0; c < NCHsz; ++c) {
    const int j0 = c * JCsz;
    tri_d1_kernel<<<dim3(193, 8, Bsz), blk, 0, stream>>>(midb, Wtri, Tc, j0);
    tri_d2_kernel<<<dim3(2, 8, Bsz * COLSsz), blk, 0, stream>>>(headb, Tc, Uc);
    tri_d3_kernel<<<dim3(4, 8, Bsz * Lsz), blk, 0, stream>>>(tailb, Uc, sbuf,
                                                             j0, c > 0 ? 1 : 0);
  }

  // E: mask + softmax over z (in place -> alpha)
  tri_softmax_z_kernel<<<dim3(16384), blk, 0, stream>>>(sbuf);
  // F: fused alpha@text -> relu -> @Vw + Vb
  tri_score_kernel<<<dim3(1024, Bsz * NCsz), blk, 0, stream>>>(sbuf, text, Vw,
                                                               Vb, scoreb);
  // G: pad/reshape/add, global min-max normalize, final softmax
  comb_kernel<<<dim3(512), dim3(256), 0, stream>>>(pattn, scoreb, comb);
  minmax_kernel<<<dim3(1), dim3(256), 0, stream>>>(comb, mm);
  final_softmax_kernel<<<dim3(256), blk, 0, stream>>>(comb, mm, (float*)d_out);
}